// MelMeanFilteredMSE_16406775071361
// MI455X (gfx1250) — compile-verified
//
#include <hip/hip_runtime.h>
#include <hip/hip_bf16.h>

typedef __attribute__((ext_vector_type(2))) float v2f;
typedef __attribute__((ext_vector_type(8))) float v8f;

#define NFREQ    1025
#define NTIME    2048
#define NBATCH   16
#define HALFBAND 10
#define ROWS     64        // output freq rows per workgroup (4 x 16-row WMMA tiles)
#define RTILES   4
#define BROWS    (ROWS + 2*HALFBAND)   // 84 diff rows staged (halo factor 1.31x)
#define KSPAN    36        // band window per 16-row tile: 16 + 2*10
#define TBLOCK   128       // time columns per workgroup (8 waves x 16)
#define YTILES   17        // ceil(1025/64)
#define LDSB_STRIDE 136    // 128+8: rows K and K+2 hit disjoint LDS bank sets
#define LDSA_STRIDE 41     // 36+5: near conflict-free A-operand reads

#define EXP10_DIV10_LOG2 0.33219280948873623f   // log2(10)/10

// wave32 XOR-butterfly add via ds_swizzle (group-of-32: offset={xor<<10 | and 0x1f})
template <int XORMASK>
__device__ __forceinline__ float swz_add(float x) {
    int v = __builtin_amdgcn_ds_swizzle(__float_as_int(x), (XORMASK << 10) | 0x1f);
    return x + __int_as_float(v);
}

// ---------------------------------------------------------------------------
// Fused: exp2 transform -> LDS stage (float4 / b128 path) -> banded GEMM via
// v_wmma_f32_16x16x4_f32 (4 row-tiles x 9 K-steps per wave) -> sum of squares.
// ---------------------------------------------------------------------------
__global__ __launch_bounds__(256)
void mel_mse_wmma_kernel(const float* __restrict__ mo,
                         const float* __restrict__ tg,
                         const float* __restrict__ tm,
                         float* __restrict__ partials)
{
    __shared__ float ldsB[BROWS * LDSB_STRIDE];          // diff window 84 x 128
    __shared__ float ldsA[RTILES * 16 * LDSA_STRIDE];    // 4 band tiles 16 x 36
    __shared__ float wsum[8];

    const int tid    = threadIdx.x;
    const int b      = blockIdx.z;
    const int i0     = blockIdx.y * ROWS;     // first output freq row of block
    const int tbase  = blockIdx.x * TBLOCK;   // first time column
    const int jstart = i0 - HALFBAND;         // first staged diff row (may be <0)

    // ---- Stage B: diff = 2^(mo*c) - 2^(tg*c), vectorized b128, zero-padded ----
    const size_t plane = (size_t)b * NFREQ * NTIME;
    for (int idx = tid; idx < BROWS * (TBLOCK / 4); idx += 256) {
        const int r  = idx >> 5;              // 0..83  (32 float4 per row)
        const int c4 = idx & 31;
        const int j  = jstart + r;
        float4 v = make_float4(0.f, 0.f, 0.f, 0.f);
        if (j >= 0 && j < NFREQ) {
            const size_t off = plane + (size_t)j * NTIME + (size_t)tbase;
            const float4 m4 = *(const float4*)(mo + off + 4 * c4);
            const float4 t4 = *(const float4*)(tg + off + 4 * c4);
            v.x = __builtin_amdgcn_exp2f(m4.x * EXP10_DIV10_LOG2) -
                  __builtin_amdgcn_exp2f(t4.x * EXP10_DIV10_LOG2);
            v.y = __builtin_amdgcn_exp2f(m4.y * EXP10_DIV10_LOG2) -
                  __builtin_amdgcn_exp2f(t4.y * EXP10_DIV10_LOG2);
            v.z = __builtin_amdgcn_exp2f(m4.z * EXP10_DIV10_LOG2) -
                  __builtin_amdgcn_exp2f(t4.z * EXP10_DIV10_LOG2);
            v.w = __builtin_amdgcn_exp2f(m4.w * EXP10_DIV10_LOG2) -
                  __builtin_amdgcn_exp2f(t4.w * EXP10_DIV10_LOG2);
        }
        *(float4*)&ldsB[r * LDSB_STRIDE + 4 * c4] = v;   // 544B row stride: 16B aligned
    }

    // ---- Stage A: per row-tile band of M, tm[i][i_tile-10 .. i_tile+25], padded ----
    for (int idx = tid; idx < RTILES * 16 * KSPAN; idx += 256) {
        const int tt = idx / (16 * KSPAN);
        const int rem = idx - tt * (16 * KSPAN);
        const int r  = rem / KSPAN;
        const int c  = rem - r * KSPAN;
        const int i  = i0 + tt * 16 + r;
        const int j  = (i0 + tt * 16 - HALFBAND) + c;
        float v = 0.0f;
        if (i < NFREQ && j >= 0 && j < NFREQ)
            v = tm[(size_t)i * NFREQ + j];
        ldsA[(tt * 16 + r) * LDSA_STRIDE + c] = v;
    }
    __syncthreads();

    // ---- Banded GEMM: 4 tiles x 9 x V_WMMA_F32_16X16X4_F32, f32-exact ----
    const int wave = tid >> 5;
    const int lane = tid & 31;
    const int ln   = lane & 15;               // M row (A) / N col (B)
    const int koff = (lane >> 4) << 1;        // 0 for lanes 0-15, 2 for lanes 16-31
    const int woff = wave * 16;               // this wave's time sub-tile

    float s = 0.f;
    #pragma unroll
    for (int tt = 0; tt < RTILES; ++tt) {
        v8f acc = {0.f, 0.f, 0.f, 0.f, 0.f, 0.f, 0.f, 0.f};
        const int arow = (tt * 16 + ln) * LDSA_STRIDE;
        const int brow = tt * 16;             // tile's band starts 16 rows later each
        #pragma unroll
        for (int k0 = 0; k0 < KSPAN; k0 += 4) {
            const int kx = k0 + koff;
            v2f a, bm;
            // A 16x4 layout (ISA 7.12.2): VGPR0={K0|K2}, VGPR1={K1|K3}, M=lane%16
            a.x = ldsA[arow + kx];
            a.y = ldsA[arow + kx + 1];
            // B 4x16 layout: row striped across lanes; VGPR v = {K=v | K=v+2}
            bm.x = ldsB[(brow + kx) * LDSB_STRIDE + woff + ln];
            bm.y = ldsB[(brow + kx + 1) * LDSB_STRIDE + woff + ln];
            acc = __builtin_amdgcn_wmma_f32_16x16x4_f32(
                      false, a, false, bm, (short)0, acc, false, false);
        }
        #pragma unroll
        for (int i = 0; i < 8; ++i) s += acc[i] * acc[i];
    }

    // ---- wave32 butterfly reduce (ds_swizzle), then block partial ----
    s = swz_add<16>(s);
    s = swz_add<8>(s);
    s = swz_add<4>(s);
    s = swz_add<2>(s);
    s = swz_add<1>(s);
    if (lane == 0) wsum[wave] = s;
    __syncthreads();
    if (tid == 0) {
        float t = 0.f;
        #pragma unroll
        for (int w = 0; w < 8; ++w) t += wsum[w];
        partials[((size_t)blockIdx.z * gridDim.y + blockIdx.y) * gridDim.x + blockIdx.x] = t;
    }
}

// ---------------------------------------------------------------------------
// Deterministic final reduction (single block, fixed-order tree).
// ---------------------------------------------------------------------------
__global__ __launch_bounds__(256)
void mel_mse_reduce_kernel(const float* __restrict__ partials, int n,
                           float* __restrict__ out, float scale)
{
    __shared__ float sm[256];
    float s = 0.f;
    for (int i = threadIdx.x; i < n; i += 256) s += partials[i];
    sm[threadIdx.x] = s;
    __syncthreads();
    for (int off = 128; off > 0; off >>= 1) {
        if ((int)threadIdx.x < off) sm[threadIdx.x] += sm[threadIdx.x + off];
        __syncthreads();
    }
    if (threadIdx.x == 0) out[0] = sm[0] * scale;
}

extern "C" void kernel_launch(void* const* d_in, const int* in_sizes, int n_in,
                              void* d_out, int out_size, void* d_ws, size_t ws_size,
                              hipStream_t stream)
{
    (void)in_sizes; (void)n_in; (void)out_size; (void)ws_size;
    const float* mo = (const float*)d_in[0];   // model_output [16,1025,2048] f32
    const float* tg = (const float*)d_in[1];   // target       [16,1025,2048] f32
    const float* tm = (const float*)d_in[2];   // transform    [1025,1025]    f32
    float* out      = (float*)d_out;           // scalar f32
    float* partials = (float*)d_ws;            // 16*17*16 = 4352 floats (~17 KB)

    dim3 grid(NTIME / TBLOCK, YTILES, NBATCH);   // (16, 17, 16) = 4352 blocks
    mel_mse_wmma_kernel<<<grid, 256, 0, stream>>>(mo, tg, tm, partials);

    const int nPart = (NTIME / TBLOCK) * YTILES * NBATCH;
    const float scale = 1.0f / ((float)NBATCH * (float)NFREQ * (float)NTIME);
    mel_mse_reduce_kernel<<<1, 256, 0, stream>>>(partials, nPart, out, scale);
}